// HOR_16870631539538
// MI455X (gfx1250) — compile-verified
//
#include <hip/hip_runtime.h>
#include <math.h>

// ---------------------------------------------------------------------------
// MI455X (gfx1250) implementation, round 3.
//  * All GEMMs on V_WMMA_F32_16X16X4_F32 (native fp32 WMMA, wave32).
//  * [hw x hw] attention matrix p never touches HBM (column softmax stats +
//    fused recompute/normalize/accumulate pass through a private LDS slice).
//  * Compile-time strides -> merged b64 fragment loads / b128 tile stores.
//  * Explicit load-phase/compute-phase + double buffering (unrolled x2 so all
//    register-array indices are compile-time): loads issue as clauses and fly
//    under the previous tile's WMMA chain instead of load->wait->wmma lockstep.
// ---------------------------------------------------------------------------

typedef __attribute__((ext_vector_type(2))) float v2f;
typedef __attribute__((ext_vector_type(4))) float v4f;
typedef __attribute__((ext_vector_type(8))) float v8f;

#define HWN 4096
#define CN 64
#define BN 4
#define CHW (CN * HWN)   // 262144 floats per batch
#define HWC (HWN * CN)

__device__ __forceinline__ v8f wmma4(v2f a, v2f b, v8f c) {
  return __builtin_amdgcn_wmma_f32_16x16x4_f32(
      false, a, false, b, (short)0, c, false, false);
}

// A fragment (M=16,K=4): lane l -> row m0+(l&15); VGPR r -> K = k0 + r + 2*(l>>4)
template <int RS, int CS>
__device__ __forceinline__ v2f load_a(const float* __restrict__ p, int m0, int k0) {
  int l = threadIdx.x & 31;
  int m = m0 + (l & 15);
  int kb = k0 + ((l >> 4) << 1);
  if constexpr (CS == 1) {
    return *(const v2f*)(p + m * RS + kb);          // merged b64 load
  } else {
    v2f a;
    a.x = p[m * RS + kb * CS];
    a.y = p[m * RS + (kb + 1) * CS];
    return a;
  }
}

// B fragment (K=4,N=16): lane l -> col n0+(l&15); VGPR r -> K = k0 + r + 2*(l>>4)
template <int RS, int CS>
__device__ __forceinline__ v2f load_b(const float* __restrict__ p, int k0, int n0) {
  int l = threadIdx.x & 31;
  int n = n0 + (l & 15);
  int kb = k0 + ((l >> 4) << 1);
  if constexpr (RS == 1) {
    return *(const v2f*)(p + kb + n * CS);          // merged b64 load
  } else {
    v2f b;
    b.x = p[kb * RS + n * CS];
    b.y = p[(kb + 1) * RS + n * CS];
    return b;
  }
}

// D fragment (16x16): lane l -> col n0+(l&15); VGPR r -> row m0 + r + 8*(l>>4)
template <int RS, int CS>
__device__ __forceinline__ void store_d(float* __restrict__ p, int m0, int n0, v8f d) {
  int l = threadIdx.x & 31;
  int n = n0 + (l & 15);
  int mb = m0 + ((l >> 4) << 3);
  if constexpr (RS == 1) {
    float* q = p + mb + n * CS;                     // 8 consecutive floats
    v4f lo = {d[0], d[1], d[2], d[3]};
    v4f hi = {d[4], d[5], d[6], d[7]};
    *(v4f*)q = lo;
    *(v4f*)(q + 4) = hi;
  } else {
#pragma unroll
    for (int r = 0; r < 8; ++r) p[(mb + r) * RS + n * CS] = d[r];
  }
}

// ---- bulk fragment loaders (whole 16-deep K chain at once; clause-friendly)
template <int RS, int CS>
__device__ __forceinline__ void load_a16(v2f (&dst)[16], const float* __restrict__ p, int m0) {
#pragma unroll
  for (int kk = 0; kk < 16; ++kk) dst[kk] = load_a<RS, CS>(p, m0, kk * 4);
}
template <int RS, int CS>
__device__ __forceinline__ void load_b16(v2f (&dst)[16], const float* __restrict__ p, int n0) {
#pragma unroll
  for (int kk = 0; kk < 16; ++kk) dst[kk] = load_b<RS, CS>(p, kk * 4, n0);
}

__device__ __forceinline__ v8f ptile16(const v2f (&af)[16], const v2f (&bf)[16]) {
  v8f p = {};
#pragma unroll
  for (int kk = 0; kk < 16; ++kk) p = wmma4(af[kk], bf[kk], p);
  return p;
}

// ---------------------------------------------------------------------------
// Generic WMMA GEMM, compile-time strides: D[m,n] = sum_k A[m,k]*B[k,n](+bias[m])
// 128 threads = 4 waves; each wave owns a 16m x 64n tile. K chunked by 16 and
// double-buffered (requires K % 32 == 0, true for all launches: 64 / 4096).
// Grid: (N/64, M/64, BN).
// ---------------------------------------------------------------------------
template <int ARS, int ACS, int BRS, int BCS>
__device__ __forceinline__ void gemm_load_chunk(v2f (&af)[4], v2f (&bf)[16],
                                                const float* __restrict__ Ap,
                                                const float* __restrict__ Bp,
                                                int m0, int n0, int k0) {
#pragma unroll
  for (int i = 0; i < 4; ++i) af[i] = load_a<ARS, ACS>(Ap, m0, k0 + 4 * i);
#pragma unroll
  for (int i = 0; i < 4; ++i)
#pragma unroll
    for (int t = 0; t < 4; ++t)
      bf[i * 4 + t] = load_b<BRS, BCS>(Bp, k0 + 4 * i, n0 + 16 * t);
}

__device__ __forceinline__ void gemm_compute(v8f (&acc)[4], const v2f (&af)[4],
                                             const v2f (&bf)[16]) {
#pragma unroll
  for (int i = 0; i < 4; ++i)
#pragma unroll
    for (int t = 0; t < 4; ++t) acc[t] = wmma4(af[i], bf[i * 4 + t], acc[t]);
}

template <int ARS, int ACS, int BRS, int BCS, int DRS, int DCS>
__global__ void wmma_gemm_bias(const float* __restrict__ A, long long ab,
                               const float* __restrict__ Bm, long long bb,
                               const float* __restrict__ bias,
                               float* __restrict__ D, long long db,
                               int M, int N, int K) {
  int wave = (int)(threadIdx.x >> 5);
  int m0 = blockIdx.y * 64 + wave * 16;
  int n0 = blockIdx.x * 64;
  int z = blockIdx.z;
  if (m0 >= M) return;  // wave-uniform: EXEC stays all-ones for WMMA
  const float* Ap = A + (long long)z * ab;
  const float* Bp = Bm + (long long)z * bb;
  float* Dp = D + (long long)z * db;

  v2f af0[4], af1[4];
  v2f bf0[16], bf1[16];
  v8f acc[4] = {};
  gemm_load_chunk<ARS, ACS, BRS, BCS>(af0, bf0, Ap, Bp, m0, n0, 0);
  for (int k0 = 0; k0 < K; k0 += 32) {
    gemm_load_chunk<ARS, ACS, BRS, BCS>(af1, bf1, Ap, Bp, m0, n0, k0 + 16);
    gemm_compute(acc, af0, bf0);
    if (k0 + 32 < K)
      gemm_load_chunk<ARS, ACS, BRS, BCS>(af0, bf0, Ap, Bp, m0, n0, k0 + 32);
    gemm_compute(acc, af1, bf1);
  }
  if (bias) {
    int l = threadIdx.x & 31;
    int mb = m0 + ((l >> 4) << 3);
#pragma unroll
    for (int t = 0; t < 4; ++t)
#pragma unroll
      for (int r = 0; r < 8; ++r) acc[t][r] += bias[mb + r];
  }
#pragma unroll
  for (int t = 0; t < 4; ++t) store_d<DRS, DCS>(Dp, m0, n0 + 16 * t, acc[t]);
}

// ---------------------------------------------------------------------------
// Column softmax stats of p[n,m] = sum_c x_low[n,c] * xl_hi[m,c]
// (softmax over n => per-column m max & exp-sum). One wave per 16 columns;
// xl_hi B-fragments hoisted; x_low A-fragments double-buffered across n tiles.
// ---------------------------------------------------------------------------
__global__ void colstats_kernel(const float* __restrict__ xlow,   // [B][HW][C]
                                const float* __restrict__ xhi,    // [B][HW][C]
                                float* __restrict__ cmax,
                                float* __restrict__ csum) {
  int z = blockIdx.y;
  int wave = (int)(threadIdx.x >> 5);
  int m0 = (blockIdx.x * 4 + wave) * 16;
  int l = threadIdx.x & 31;
  const float* XL = xlow + (long long)z * HWC;
  const float* XH = xhi + (long long)z * HWC;

  v2f bfrag[16];  // B[k=c, m] = xl_hi[m, c]  -> RS=1, CS=64
  load_b16<1, CN>(bfrag, XH, m0);

  v2f af0[16], af1[16];
  float vmax = -INFINITY;
  load_a16<CN, 1>(af0, XL, 0);
  for (int n0 = 0; n0 < HWN; n0 += 32) {
    __builtin_prefetch(XL + (n0 + 32) * CN + l * 32, 0, 3);
    load_a16<CN, 1>(af1, XL, n0 + 16);
    v8f p = ptile16(af0, bfrag);
#pragma unroll
    for (int r = 0; r < 8; ++r) vmax = fmaxf(vmax, p[r]);
    if (n0 + 32 < HWN) load_a16<CN, 1>(af0, XL, n0 + 32);
    p = ptile16(af1, bfrag);
#pragma unroll
    for (int r = 0; r < 8; ++r) vmax = fmaxf(vmax, p[r]);
  }
  vmax = fmaxf(vmax, __shfl_xor(vmax, 16, 32));  // combine lane halves (same column)

  float vsum = 0.0f;
  load_a16<CN, 1>(af0, XL, 0);
  for (int n0 = 0; n0 < HWN; n0 += 32) {
    load_a16<CN, 1>(af1, XL, n0 + 16);
    v8f p = ptile16(af0, bfrag);
#pragma unroll
    for (int r = 0; r < 8; ++r) vsum += __expf(p[r] - vmax);
    if (n0 + 32 < HWN) load_a16<CN, 1>(af0, XL, n0 + 32);
    p = ptile16(af1, bfrag);
#pragma unroll
    for (int r = 0; r < 8; ++r) vsum += __expf(p[r] - vmax);
  }
  vsum += __shfl_xor(vsum, 16, 32);

  if (l < 16) {
    cmax[(long long)z * HWN + m0 + l] = vmax;
    csum[(long long)z * HWN + m0 + l] = vsum;
  }
}

// ---------------------------------------------------------------------------
// e_pre[n,c] = sum_m softmax_col(p)[n,m] * v[m,c] + x_low[n,c]   ([B][HW][C])
// Per wave: 16 n-rows x full c=64 in 4 accumulators. Per m-tile:
// WMMA p-tile -> (v loads issued, fly under...) exp/normalize -> stage in the
// wave's private LDS slice (D->A layout; LDS is in-order within a wave, no
// barrier needed) -> WMMA-accumulate against v.
// ---------------------------------------------------------------------------
__device__ __forceinline__ void epre_tile(int m0, int l,
                                          const v2f (&afrag)[16], const v2f (&bh)[16],
                                          const float* __restrict__ V,
                                          const float* __restrict__ CM,
                                          const float* __restrict__ CS,
                                          float (&lds)[16][18], v8f (&acc)[4]) {
  // p tile [16n x 16m]
  v8f p = ptile16(afrag, bh);
  // issue v B-fragments now; they complete under the exp/normalize + staging
  v2f vf[16];  // V[c*HW + m] -> RS=1, CS=HW
#pragma unroll
  for (int ks = 0; ks < 4; ++ks)
#pragma unroll
    for (int ct = 0; ct < 4; ++ct)
      vf[ks * 4 + ct] = load_b<1, HWN>(V, m0 + ks * 4, ct * 16);
  // column-normalize: this lane holds column m = m0 + (l&15)
  float cm = CM[m0 + (l & 15)];
  float inv = 1.0f / CS[m0 + (l & 15)];
#pragma unroll
  for (int r = 0; r < 8; ++r) p[r] = __expf(p[r] - cm) * inv;
  // stage normalized tile (row r+8*(l>>4), col l&15)
#pragma unroll
  for (int r = 0; r < 8; ++r) lds[r + ((l >> 4) << 3)][l & 15] = p[r];
  // acc[n,c] += p_norm[n, m0+k] * v[m0+k, c]
#pragma unroll
  for (int ks = 0; ks < 4; ++ks) {
    v2f a = load_a<18, 1>(&lds[0][0], 0, ks * 4);  // ds_load_b64
#pragma unroll
    for (int ct = 0; ct < 4; ++ct) acc[ct] = wmma4(a, vf[ks * 4 + ct], acc[ct]);
  }
}

__global__ void epre_kernel(const float* __restrict__ xlow,  // [B][HW][C]
                            const float* __restrict__ xhi,   // [B][HW][C]
                            const float* __restrict__ vmat,  // [B][C][HW]
                            const float* __restrict__ cmax,
                            const float* __restrict__ csum,
                            float* __restrict__ eP) {        // [B][HW][C]
  __shared__ float lds[4][16][18];  // pad 18: bank spread + aligned b64 reads
  int z = blockIdx.y;
  int wave = (int)(threadIdx.x >> 5);
  int n0 = (blockIdx.x * 4 + wave) * 16;
  int l = threadIdx.x & 31;
  const float* XL = xlow + (long long)z * HWC;
  const float* XH = xhi + (long long)z * HWC;
  const float* V = vmat + (long long)z * CHW;
  const float* CM = cmax + (long long)z * HWN;
  const float* CS = csum + (long long)z * HWN;

  v2f afrag[16];  // x_low rows n0..n0+15 over K=c, reused for every m tile
  load_a16<CN, 1>(afrag, XL, n0);

  v8f acc[4] = {};  // 16n x 64c
  v2f bh0[16], bh1[16];
  load_b16<1, CN>(bh0, XH, 0);
  for (int m0 = 0; m0 < HWN; m0 += 32) {
    __builtin_prefetch(XH + (m0 + 32) * CN + l * 32, 0, 3);
    __builtin_prefetch(V + (long long)(l * 2) * HWN + m0 + 32, 0, 3);
    load_b16<1, CN>(bh1, XH, m0 + 16);
    epre_tile(m0, l, afrag, bh0, V, CM, CS, lds[wave], acc);
    if (m0 + 32 < HWN) load_b16<1, CN>(bh0, XH, m0 + 32);
    epre_tile(m0 + 16, l, afrag, bh1, V, CM, CS, lds[wave], acc);
  }

  // residual + store: acc[ct] row n = n0 + r + 8*(l>>4), col c = ct*16 + (l&15)
  float* EP = eP + (long long)z * HWC;
  int nb = n0 + ((l >> 4) << 3);
#pragma unroll
  for (int ct = 0; ct < 4; ++ct) {
    int c = ct * 16 + (l & 15);
#pragma unroll
    for (int r = 0; r < 8; ++r)
      EP[(nb + r) * CN + c] = acc[ct][r] + XL[(nb + r) * CN + c];
  }
}

// ---------------------------------------------------------------------------
// 64x64 channel softmax over axis=1 (rows c, per column d), stored TRANSPOSED
// as tsmT[d][c] so the final GEMM's B fragment has K-stride 1.
// ---------------------------------------------------------------------------
__global__ void softmax_t_kernel(const float* __restrict__ t, float* __restrict__ tsmT) {
  int z = blockIdx.x;
  int d = threadIdx.x;
  const float* T = t + (long long)z * CN * CN;
  float* O = tsmT + (long long)z * CN * CN;
  float mx = -INFINITY;
  for (int c = 0; c < CN; ++c) mx = fmaxf(mx, T[c * CN + d]);
  float s = 0.0f;
  for (int c = 0; c < CN; ++c) s += __expf(T[c * CN + d] - mx);
  float inv = 1.0f / s;
  for (int c = 0; c < CN; ++c) O[d * CN + c] = __expf(T[c * CN + d] - mx) * inv;
}

// ---------------------------------------------------------------------------
extern "C" void kernel_launch(void* const* d_in, const int* in_sizes, int n_in,
                              void* d_out, int out_size, void* d_ws, size_t ws_size,
                              hipStream_t stream) {
  const float* x_latter = (const float*)d_in[0];   // [B][C][HW]
  const float* x        = (const float*)d_in[1];   // [B][C][HW]
  const float* W_high   = (const float*)d_in[2];
  const float* b_high   = (const float*)d_in[3];
  const float* W_low    = (const float*)d_in[4];
  const float* b_low    = (const float*)d_in[5];
  const float* W_value  = (const float*)d_in[6];
  const float* b_value  = (const float*)d_in[7];
  const float* W_econv  = (const float*)d_in[8];
  const float* b_econv  = (const float*)d_in[9];
  const float* W_mid    = (const float*)d_in[10];
  const float* b_mid    = (const float*)d_in[11];
  const float* W_latter = (const float*)d_in[12];
  const float* b_latter = (const float*)d_in[13];
  float* out = (float*)d_out;                       // [B][C][HW]

  // workspace carve-up (~30 MB, fp32)
  float* ws    = (float*)d_ws;
  float* xl_hi = ws;                                // [B][HW][C]
  float* x_low = xl_hi + (long long)BN * HWC;       // [B][HW][C]
  float* vbuf  = x_low + (long long)BN * HWC;       // [B][C][HW]
  float* xl2   = vbuf  + (long long)BN * CHW;       // [B][C][HW]
  float* x_mid = xl2   + (long long)BN * CHW;       // [B][HW][C]
  float* ePre  = x_mid + (long long)BN * HWC;       // [B][HW][C]
  float* ebuf  = ePre  + (long long)BN * HWC;       // [B][C][HW]
  float* tbuf  = ebuf  + (long long)BN * CHW;       // [B][64][64] (c,d)
  float* tsmT  = tbuf  + (long long)BN * CN * CN;   // [B][64][64] (d,c)
  float* cmax  = tsmT  + (long long)BN * CN * CN;   // [B][HW]
  float* csum  = cmax  + (long long)BN * HWN;

  dim3 blk(128);

  // --- stage 0: five front 1x1 convs, D[o,n] = sum_c W[o,c]*src[c,n] + b[o]
  {
    dim3 g(HWN / 64, 1, BN);
    wmma_gemm_bias<CN, 1, HWN, 1, 1, CN><<<g, blk, 0, stream>>>(
        W_high, 0, x_latter, CHW, b_high, xl_hi, HWC, CN, HWN, CN);
    wmma_gemm_bias<CN, 1, HWN, 1, 1, CN><<<g, blk, 0, stream>>>(
        W_low, 0, x, CHW, b_low, x_low, HWC, CN, HWN, CN);
    wmma_gemm_bias<CN, 1, HWN, 1, 1, CN><<<g, blk, 0, stream>>>(
        W_mid, 0, x, CHW, b_mid, x_mid, HWC, CN, HWN, CN);
    wmma_gemm_bias<CN, 1, HWN, 1, HWN, 1><<<g, blk, 0, stream>>>(
        W_value, 0, x, CHW, b_value, vbuf, CHW, CN, HWN, CN);
    wmma_gemm_bias<CN, 1, HWN, 1, HWN, 1><<<g, blk, 0, stream>>>(
        W_latter, 0, x_latter, CHW, b_latter, xl2, CHW, CN, HWN, CN);
  }

  // --- stage 1: column softmax stats of p (p never hits HBM)
  {
    dim3 g(HWN / 64, BN);
    colstats_kernel<<<g, blk, 0, stream>>>(x_low, xl_hi, cmax, csum);
  }

  // --- stage 1b: e_pre = softmax_col(p) @ v + x_low   (fused, streamed)
  {
    dim3 g(HWN / 64, BN);
    epre_kernel<<<g, blk, 0, stream>>>(x_low, xl_hi, vbuf, cmax, csum, ePre);
  }

  // --- e = e_conv(e_pre): e[o,n] = sum_c We[o,c]*e_pre[n,c] + be[o] -> [C][HW]
  {
    dim3 g(HWN / 64, 1, BN);
    wmma_gemm_bias<CN, 1, 1, CN, HWN, 1><<<g, blk, 0, stream>>>(
        W_econv, 0, ePre, HWC, b_econv, ebuf, CHW, CN, HWN, CN);
  }

  // --- stage 2: t[c,d] = sum_n e[c,n]*xl2[d,n]^T   (64x64, K=4096)
  {
    dim3 g(1, 1, BN);
    wmma_gemm_bias<HWN, 1, 1, HWN, CN, 1><<<g, blk, 0, stream>>>(
        ebuf, CHW, xl2, CHW, nullptr, tbuf, (long long)CN * CN, CN, CN, HWN);
  }

  // --- softmax over axis=1 (rows c per column d), store transposed (d,c)
  softmax_t_kernel<<<BN, CN, 0, stream>>>(tbuf, tsmT);

  // --- out[n,d] = sum_c x_mid[n,c]*tsmT[d,c]; store as [d][n] in d_out
  {
    dim3 g(1, HWN / 64, BN);
    wmma_gemm_bias<CN, 1, 1, CN, 1, HWN><<<g, blk, 0, stream>>>(
        x_mid, HWC, tsmT, (long long)CN * CN, nullptr, out, CHW, HWN, CN, CN);
  }
}